// ModalGATLayer_47519518163349
// MI455X (gfx1250) — compile-verified
//
#include <hip/hip_runtime.h>

#define N_NODES 100000
#define N_EDGES 1200000
#define DIM 64
#define TOT_E (N_EDGES + N_NODES)

typedef float v2f __attribute__((ext_vector_type(2)));
typedef float v8f __attribute__((ext_vector_type(8)));

// ---------------- init: zero accumulators ----------------
__global__ void k_init(float* __restrict__ out_acc, float* __restrict__ denom,
                       unsigned* __restrict__ emax_key) {
    int tid = blockIdx.x * blockDim.x + threadIdx.x;
    if (tid < N_NODES * DIM) out_acc[tid] = 0.f;
    if (tid < N_NODES) { denom[tid] = 0.f; emax_key[tid] = 0u; }
}

// -------- GEMM h = x@W with fused a_s = h@att_src, a_d = h@att_dst --------
// Block = 128 threads (4 waves). Block computes a 16x64 tile of h; wave w
// owns the 16x16 sub-tile at columns [16w, 16w+16). K=64 -> 16 WMMA f32
// 16x16x4 ops per wave.
__global__ __launch_bounds__(128)
void k_gemm(const float* __restrict__ x, const float* __restrict__ W,
            const float* __restrict__ att_src, const float* __restrict__ att_dst,
            float* __restrict__ h, float* __restrict__ a_s, float* __restrict__ a_d) {
    __shared__ float Wl[64 * 65];   // padded stride 65 -> conflict-free halves
    __shared__ float sAS[4][16];
    __shared__ float sAD[4][16];

    const int tid  = threadIdx.x;
    const int wave = tid >> 5;
    const int lane = tid & 31;
    const int half = lane >> 4;     // 0: lanes 0-15, 1: lanes 16-31
    const int l16  = lane & 15;
    const int m0   = blockIdx.x * 16;
    const int n0   = wave * 16;

    // stage W (row-major [k][n]) into LDS
    for (int i = tid; i < 64 * 64; i += 128) {
        int k = i >> 6, n = i & 63;
        Wl[k * 65 + n] = W[i];
    }
    __syncthreads();

    // A 16x4 f32 layout: lanes 0-15 hold M=lane, K = k0,k0+1; lanes 16-31
    // hold M=lane-16, K = k0+2,k0+3.  B 4x16 symmetric over N.
    v8f acc = {};
    const int arow = m0 + l16;
    const int nB   = n0 + l16;
    #pragma unroll
    for (int k0 = 0; k0 < 64; k0 += 4) {
        const int kA = k0 + half * 2;
        v2f a = *(const v2f*)(x + (size_t)arow * DIM + kA);
        v2f b;
        b.x = Wl[kA * 65 + nB];
        b.y = Wl[(kA + 1) * 65 + nB];
        acc = __builtin_amdgcn_wmma_f32_16x16x4_f32(false, a, false, b,
                                                    (short)0, acc, false, false);
    }

    // D layout: VGPR r -> row m0 + r + half*8, col n0 + l16
    const int col = n0 + l16;
    #pragma unroll
    for (int r = 0; r < 8; ++r) {
        int row = m0 + r + half * 8;
        h[(size_t)row * DIM + col] = acc[r];
    }

    // fused attention dot products: reduce each row over this wave's 16 cols
    const float as_c = att_src[col];
    const float ad_c = att_dst[col];
    #pragma unroll
    for (int r = 0; r < 8; ++r) {
        float vs = acc[r] * as_c;
        float vd = acc[r] * ad_c;
        #pragma unroll
        for (int off = 8; off >= 1; off >>= 1) {
            vs += __shfl_xor(vs, off, 32);
            vd += __shfl_xor(vd, off, 32);
        }
        if (l16 == 0) {
            sAS[wave][r + half * 8] = vs;
            sAD[wave][r + half * 8] = vd;
        }
    }
    __syncthreads();
    // combine the 4 column-tiles
    if (tid < 16) {
        float s = 0.f, d = 0.f;
        #pragma unroll
        for (int w2 = 0; w2 < 4; ++w2) { s += sAS[w2][tid]; d += sAD[w2][tid]; }
        a_s[m0 + tid] = s;
        a_d[m0 + tid] = d;
    }
}

// order-preserving float<->uint mapping for atomic max
__device__ __forceinline__ unsigned f2key(float f) {
    unsigned u = __float_as_uint(f);
    return (u & 0x80000000u) ? ~u : (u | 0x80000000u);
}
__device__ __forceinline__ float key2f(unsigned k) {
    unsigned u = (k & 0x80000000u) ? (k ^ 0x80000000u) : ~k;
    return __uint_as_float(u);
}

__device__ __forceinline__ void edge_ids(const long long* __restrict__ ei, int i,
                                         int& src, int& dst) {
    if (i < N_EDGES) {
        src = (int)ei[i];
        dst = (int)ei[N_EDGES + i];
    } else {
        src = dst = i - N_EDGES;   // self loops appended after real edges
    }
}

// ---- pass A: per-edge leaky-relu score + segment max over dst ----
__global__ void k_score(const long long* __restrict__ ei,
                        const float* __restrict__ a_s, const float* __restrict__ a_d,
                        float* __restrict__ esc, unsigned* __restrict__ emax_key) {
    int i = blockIdx.x * blockDim.x + threadIdx.x;
    if (i >= TOT_E) return;
    int src, dst;
    edge_ids(ei, i, src, dst);
    float e = a_s[src] + a_d[dst];
    e = (e > 0.f) ? e : 0.2f * e;           // leaky_relu slope 0.2
    esc[i] = e;
    atomicMax(&emax_key[dst], f2key(e));
}

// ---- pass B: exp(e - max) + segment sum over dst ----
__global__ void k_expsum(const long long* __restrict__ ei,
                         const unsigned* __restrict__ emax_key,
                         float* __restrict__ esc, float* __restrict__ denom) {
    int i = blockIdx.x * blockDim.x + threadIdx.x;
    if (i >= TOT_E) return;
    int src, dst;
    edge_ids(ei, i, src, dst);
    float ex = __expf(esc[i] - key2f(emax_key[dst]));
    esc[i] = ex;
    atomicAdd(&denom[dst], ex);
}

// ---- pass C: scatter alpha * h[src] into out. 16 lanes per edge, 4 feats/lane ----
__global__ void k_aggregate(const long long* __restrict__ ei,
                            const float* __restrict__ esc,
                            const float* __restrict__ denom,
                            const float* __restrict__ h,
                            float* __restrict__ out) {
    long long t = (long long)blockIdx.x * blockDim.x + threadIdx.x;
    int edge = (int)(t >> 4);
    int seg  = (int)(t & 15);
    if (edge >= TOT_E) return;
    int src, dst;
    edge_ids(ei, edge, src, dst);
    float alpha = esc[edge] / denom[dst];
    const float4 hv = *(const float4*)(h + (size_t)src * DIM + seg * 4);
    float* op = out + (size_t)dst * DIM + seg * 4;
    atomicAdd(op + 0, alpha * hv.x);
    atomicAdd(op + 1, alpha * hv.y);
    atomicAdd(op + 2, alpha * hv.z);
    atomicAdd(op + 3, alpha * hv.w);
}

// ---- bias + LayerNorm, one wave32 per node (2 features per lane) ----
__global__ __launch_bounds__(256)
void k_layernorm(float* __restrict__ out, const float* __restrict__ bias,
                 const float* __restrict__ gamma, const float* __restrict__ beta) {
    int wave = threadIdx.x >> 5;
    int lane = threadIdx.x & 31;
    int node = blockIdx.x * 8 + wave;
    if (node >= N_NODES) return;
    const int f0 = lane, f1 = lane + 32;
    float* row = out + (size_t)node * DIM;
    float v0 = row[f0] + bias[f0];
    float v1 = row[f1] + bias[f1];
    float s1 = v0 + v1;
    float s2 = v0 * v0 + v1 * v1;
    #pragma unroll
    for (int off = 16; off >= 1; off >>= 1) {
        s1 += __shfl_xor(s1, off, 32);
        s2 += __shfl_xor(s2, off, 32);
    }
    float mean = s1 * (1.f / 64.f);
    float var  = s2 * (1.f / 64.f) - mean * mean;
    float rstd = rsqrtf(var + 1e-5f);
    row[f0] = (v0 - mean) * rstd * gamma[f0] + beta[f0];
    row[f1] = (v1 - mean) * rstd * gamma[f1] + beta[f1];
}

extern "C" void kernel_launch(void* const* d_in, const int* in_sizes, int n_in,
                              void* d_out, int out_size, void* d_ws, size_t ws_size,
                              hipStream_t stream) {
    const float*     x       = (const float*)d_in[0];
    const long long* ei      = (const long long*)d_in[1];   // int64 [2, E]
    const float*     W       = (const float*)d_in[2];
    const float*     att_src = (const float*)d_in[3];
    const float*     att_dst = (const float*)d_in[4];
    const float*     bias    = (const float*)d_in[5];
    const float*     gamma   = (const float*)d_in[6];
    const float*     beta    = (const float*)d_in[7];
    float* out = (float*)d_out;

    // workspace carve
    float*    h     = (float*)d_ws;                       // N*64
    float*    a_s   = h + (size_t)N_NODES * DIM;          // N
    float*    a_d   = a_s + N_NODES;                      // N
    unsigned* emax  = (unsigned*)(a_d + N_NODES);         // N
    float*    denom = (float*)(emax + N_NODES);           // N
    float*    esc   = denom + N_NODES;                    // E+N

    // 0) zero d_out / denom / emax
    k_init<<<(N_NODES * DIM + 255) / 256, 256, 0, stream>>>(out, denom, emax);
    // 1) h = x@W (WMMA), fused a_s/a_d
    k_gemm<<<N_NODES / 16, 128, 0, stream>>>(x, W, att_src, att_dst, h, a_s, a_d);
    // 2) edge scores + segment max
    k_score<<<(TOT_E + 255) / 256, 256, 0, stream>>>(ei, a_s, a_d, esc, emax);
    // 3) exp + segment sum
    k_expsum<<<(TOT_E + 255) / 256, 256, 0, stream>>>(ei, emax, esc, denom);
    // 4) scatter-aggregate alpha * h[src]
    {
        long long threads = (long long)TOT_E * 16;
        int blocks = (int)((threads + 255) / 256);
        k_aggregate<<<blocks, 256, 0, stream>>>(ei, esc, denom, h, out);
    }
    // 5) bias + LayerNorm (in place on d_out)
    k_layernorm<<<(N_NODES + 7) / 8, 256, 0, stream>>>(out, bias, gamma, beta);
}